// MultiHashingLayerDropout_79448305042059
// MI455X (gfx1250) — compile-verified
//
#include <hip/hip_runtime.h>
#include <hip/hip_bf16.h>

// Multi-hash embedding gather-sum:
//   out[b,s,:] = sum_h p[ids[b,s], h] * W[ ids!=0 ? hash_tables[ids[b,s],h] : 0, : ]
//
// Memory-bound (AI ~0.33 flop/byte, ~320 MB -> ~13.7 us floor @ 23.3 TB/s).
// gfx1250 strategy:
//  - wave32: four tokens per wave; 128 floats / 32 lanes = one B128 per lane/row
//  - wave-uniform index chain -> wide s_load_b128/b64 through the scalar cache;
//    all 8 pair-loads issued before any consumer (single kmcnt drain), then
//    branchless s_cselect for the id==0 zero-row case
//  - global_prefetch_b8 of all gathered rows before the consume loop
//  - non-temporal B128 stores for the streamed output (preserve L2 for W,
//    which has ~32% repeat-bucket reuse and mostly fits in 192 MB L2)
//  - remainder tokens handled by a separate tiny kernel, launched only when
//    n_tok % 32 != 0 (never for B*S = 204800) -> hot kernel is guard-free

#define EMBED           128
#define TOK_PER_WAVE    4
#define WAVES_PER_BLOCK 8
#define BLOCK_THREADS   (WAVES_PER_BLOCK * 32)
#define TOK_PER_BLOCK   (WAVES_PER_BLOCK * TOK_PER_WAVE)

typedef float float4v __attribute__((ext_vector_type(4)));
typedef float float2v __attribute__((ext_vector_type(2)));
typedef int   int4v   __attribute__((ext_vector_type(4)));
typedef int   int2v   __attribute__((ext_vector_type(2)));

__global__ __launch_bounds__(BLOCK_THREADS) void
mh_embed_kernel(const int* __restrict__ ids,        // [B*S]
                const float* __restrict__ W,        // [NB+1, 128]
                const float* __restrict__ p,        // [V, 2]
                const int* __restrict__ ht,         // [V, 2]
                float* __restrict__ out)            // [B*S, 128]
{
    const int lane = (int)(threadIdx.x & 31);
    // Force the wave id into an SGPR so every per-token address below is
    // provably wave-uniform -> scalar-cache loads, zero VMEM for indices.
    const int wave = __builtin_amdgcn_readfirstlane((int)(threadIdx.x >> 5));
    const int tok0 = ((int)blockIdx.x * WAVES_PER_BLOCK + wave) * TOK_PER_WAVE;
    const int cofs = lane * 4;               // this lane's float4 within a row

    // 4 consecutive token ids: one s_load_b128 (16B aligned: tok0 % 4 == 0).
    const int4v idv = *(const int4v*)(ids + tok0);

    // Issue ALL index/weight pair loads first (8x s_load_b64, addresses depend
    // only on idv) so they overlap under a single kmcnt drain.
    int2v   bv[TOK_PER_WAVE];
    float2v wv[TOK_PER_WAVE];
#pragma unroll
    for (int t = 0; t < TOK_PER_WAVE; ++t) {
        const int id = idv[t];
        bv[t] = *(const int2v*)(ht + 2 * id);    // ht[0..1] valid: load always
        wv[t] = *(const float2v*)(p + 2 * id);
    }

    // Branchless bucket select: id==0 -> row 0 of W (the all-zero row).
    int bkt0[TOK_PER_WAVE], bkt1[TOK_PER_WAVE];
#pragma unroll
    for (int t = 0; t < TOK_PER_WAVE; ++t) {
        const bool nz = (idv[t] != 0);
        bkt0[t] = nz ? bv[t][0] : 0;             // s_cselect
        bkt1[t] = nz ? bv[t][1] : 0;
    }

    // Prefetch every gathered row (global_prefetch_b8): gets the random
    // id -> bucket -> row-address chain off the critical path.
#pragma unroll
    for (int t = 0; t < TOK_PER_WAVE; ++t) {
        __builtin_prefetch(W + (size_t)bkt0[t] * EMBED + cofs, 0, 3);
        __builtin_prefetch(W + (size_t)bkt1[t] * EMBED + cofs, 0, 3);
    }

    // Gather (B128), weighted sum, non-temporal B128 store.
#pragma unroll
    for (int t = 0; t < TOK_PER_WAVE; ++t) {
        const float4v a = *(const float4v*)(W + (size_t)bkt0[t] * EMBED + cofs);
        const float4v b = *(const float4v*)(W + (size_t)bkt1[t] * EMBED + cofs);
        const float4v r = a * wv[t][0] + b * wv[t][1];
        __builtin_nontemporal_store(
            r, (float4v*)(out + (size_t)(tok0 + t) * EMBED + cofs));
    }
}

// Remainder kernel: one token per wave, fully guarded. Launched only when
// n_tok % TOK_PER_BLOCK != 0 (never for the 204800-token harness shape).
__global__ __launch_bounds__(1024) void
mh_embed_tail_kernel(const int* __restrict__ ids, const float* __restrict__ W,
                     const float* __restrict__ p, const int* __restrict__ ht,
                     float* __restrict__ out, int start, int n_tok)
{
    const int lane = (int)(threadIdx.x & 31);
    const int wave = __builtin_amdgcn_readfirstlane((int)(threadIdx.x >> 5));
    const int tok  = start + wave;
    if (tok >= n_tok) return;
    const int id   = ids[tok];
    const bool nz  = (id != 0);
    const int i0   = nz ? ht[2 * id + 0] : 0;
    const int i1   = nz ? ht[2 * id + 1] : 0;
    const float w0 = p[2 * id + 0];
    const float w1 = p[2 * id + 1];
    const int cofs = lane * 4;
    const float4v a = *(const float4v*)(W + (size_t)i0 * EMBED + cofs);
    const float4v b = *(const float4v*)(W + (size_t)i1 * EMBED + cofs);
    const float4v r = a * w0 + b * w1;
    __builtin_nontemporal_store(r, (float4v*)(out + (size_t)tok * EMBED + cofs));
}

extern "C" void kernel_launch(void* const* d_in, const int* in_sizes, int n_in,
                              void* d_out, int out_size, void* d_ws, size_t ws_size,
                              hipStream_t stream)
{
    // setup_inputs() order: ids (int32), W (f32), p (f32), hash_tables (int32)
    const int*   ids = (const int*)  d_in[0];
    const float* W   = (const float*)d_in[1];
    const float* p   = (const float*)d_in[2];
    const int*   ht  = (const int*)  d_in[3];
    float*       out = (float*)d_out;

    const int n_tok  = in_sizes[0];                  // B*S = 204800
    const int blocks = n_tok / TOK_PER_BLOCK;        // 6400 full blocks
    const int done   = blocks * TOK_PER_BLOCK;

    if (blocks > 0)
        mh_embed_kernel<<<blocks, BLOCK_THREADS, 0, stream>>>(ids, W, p, ht, out);
    if (done < n_tok)                                // never for 204800
        mh_embed_tail_kernel<<<1, 1024, 0, stream>>>(ids, W, p, ht, out,
                                                     done, n_tok);
}